// sGWT_29686813950006
// MI455X (gfx1250) — compile-verified
//
#include <hip/hip_runtime.h>
#include <hip/hip_bf16.h>

// Problem constants (from reference setup_inputs)
#define BATCH 8
#define NTOK  4096
#define DDIM  768
#define KSLOT 4
#define NCH   (DDIM / 4)   // 192 k-chunks of 4 for WMMA f32 16x16x4

typedef __attribute__((ext_vector_type(2))) float v2f;
typedef __attribute__((ext_vector_type(4))) float v4f;
typedef __attribute__((ext_vector_type(8))) float v8f;

// Low 32 bits of a generic pointer to a __shared__ object == LDS byte offset
// (flat LDS aperture is {SHARED_BASE[63:32], offset32}; ISA 10.2 aperture rules).
__device__ __forceinline__ unsigned lds_off(const void* p) {
    return (unsigned)(unsigned long long)p;
}

// ---------------------------------------------------------------------------
// Kernel 1: layernorm the K=4 slot vectors and build:
//   wsb  : WMMA B-operand stream, 192 chunks x 32 lanes x 2 floats (48 KB),
//          pre-swizzled into the V_WMMA_F32_16X16X4_F32 B layout:
//            lane<16 -> B[K=0..1][n=lane], lane>=16 -> B[K=2..3][n=lane-16]
//          column n<4  = a_n[d] = ln_in_w[d]*kn[n][d]
//          column n==4 = 1.0   (so the WMMA chain also yields sum_d x[d])
//          column n>4  = 0.0
//   wscal: swk[k]=sum_d w*kn  (idx 0..3),  bk[k]=sum_d b*kn (idx 4..7)
// ---------------------------------------------------------------------------
__global__ __launch_bounds__(256) void prep_kernel(
    const float* __restrict__ slots_k,
    const float* __restrict__ in_w, const float* __restrict__ in_b,
    const float* __restrict__ sl_w, const float* __restrict__ sl_b,
    float* __restrict__ wsb, float* __restrict__ wscal)
{
    __shared__ float a_s[KSLOT * DDIM];   // 12 KB
    __shared__ float r1[256], r2[256];
    const int tid = threadIdx.x;

    for (int k = 0; k < KSLOT; ++k) {
        float s = 0.f, s2 = 0.f;
        for (int d = tid; d < DDIM; d += 256) {
            float v = slots_k[k * DDIM + d];
            s += v; s2 += v * v;
        }
        r1[tid] = s; r2[tid] = s2; __syncthreads();
        for (int off = 128; off > 0; off >>= 1) {
            if (tid < off) { r1[tid] += r1[tid + off]; r2[tid] += r2[tid + off]; }
            __syncthreads();
        }
        const float mu  = r1[0] * (1.f / DDIM);
        const float var = r2[0] * (1.f / DDIM) - mu * mu;
        const float inv = rsqrtf(var + 1e-5f);
        __syncthreads();   // r1/r2 about to be reused

        float swk = 0.f, bk = 0.f;
        for (int d = tid; d < DDIM; d += 256) {
            float kn = (slots_k[k * DDIM + d] - mu) * inv * sl_w[d] + sl_b[d];
            float av = in_w[d] * kn;
            a_s[k * DDIM + d] = av;
            swk += av;
            bk  += in_b[d] * kn;
        }
        r1[tid] = swk; r2[tid] = bk; __syncthreads();
        for (int off = 128; off > 0; off >>= 1) {
            if (tid < off) { r1[tid] += r1[tid + off]; r2[tid] += r2[tid + off]; }
            __syncthreads();
        }
        if (tid == 0) { wscal[k] = r1[0]; wscal[KSLOT + k] = r2[0]; }
        __syncthreads();
    }

    // Emit the swizzled B-operand stream.
    for (int idx = tid; idx < NCH * 64; idx += 256) {
        int j  = idx >> 6;          // chunk
        int r  = idx & 63;
        int ln = r >> 1;            // lane
        int cc = r & 1;             // vgpr within lane pair
        int n  = ln & 15;           // output column (slot index / sum column)
        int dd = 4 * j + 2 * (ln >> 4) + cc;   // k-dim element within chunk
        float v = (n < KSLOT) ? a_s[n * DDIM + dd] : (n == KSLOT ? 1.f : 0.f);
        wsb[idx] = v;
    }
}

// ---------------------------------------------------------------------------
// Kernel 2: out[b][k][d] = slots_v[k][d]   (updates are atomically added later)
// ---------------------------------------------------------------------------
__global__ __launch_bounds__(256) void init_out_kernel(
    const float* __restrict__ sv, float* __restrict__ out)
{
    int i = blockIdx.x * 256 + threadIdx.x;       // 24576 total
    out[i] = sv[i % (KSLOT * DDIM)];
}

// ---------------------------------------------------------------------------
// Kernel 3: fused LN + scores(WMMA) + softmax + p^T * x accumulation.
// One wave = one 16-token tile. 256 blocks x 8 waves = 2048 tiles = 32768 tokens.
// ---------------------------------------------------------------------------
__global__ __launch_bounds__(256) void fused_kernel(
    const float* __restrict__ x, const float* __restrict__ wsb,
    const float* __restrict__ wscal, float* __restrict__ out)
{
    __shared__ float bop[NCH * 64];         // 48 KB: B-operand stream (block-shared)
    __shared__ float csc[8][16][16];        // 8 KB: per-wave C spill
    __shared__ float pbuf[8][16][KSLOT];    // 2 KB: per-wave softmax weights
    __shared__ float lacc[KSLOT * DDIM];    // 12 KB: block (K,D) accumulator
    const int tid  = threadIdx.x;
    const int wave = tid >> 5;
    const int lane = tid & 31;
    const int half = lane >> 4;             // 0/1 (WMMA lane half)
    const int lm   = lane & 15;

    // ---- Async-stage the 48 KB B-operand stream into LDS (ASYNCcnt path) ----
    // 3072 x b128 lane-copies, 12 per thread; overlaps with lacc zeroing below.
#pragma unroll
    for (int i = 0; i < (NCH * 16) / 256; ++i) {           // NCH*64/4 float4s
        const int e = (i * 256 + tid) * 4;
        const unsigned l = lds_off(&bop[e]);
        const float* g = wsb + e;
        asm volatile("global_load_async_to_lds_b128 %0, %1, off"
                     :: "v"(l), "v"(g) : "memory");
    }

    for (int i = tid; i < KSLOT * DDIM; i += 256) lacc[i] = 0.f;

    asm volatile("s_wait_asynccnt 0x0" ::: "memory");
    __syncthreads();

    const int  tile = blockIdx.x * 8 + wave;     // 0..2047
    const long tok0 = (long)tile * 16;           // flat token index (b*N + n)

    // ----- Phase 1: scores via WMMA f32 16x16x4, D = A(x) * B(a_pad) + C -----
    // A layout: lane lm holds token (tok0+lm), VGPR0/1 = k-elems 2*half, 2*half+1
    // Per chunk: 1 global b64 (A) + 1 ds b64 (B, bank-conflict-free) + 1 WMMA.
    const float* xrow = x + (tok0 + lm) * DDIM + 2 * half;
    v8f  c = {0.f, 0.f, 0.f, 0.f, 0.f, 0.f, 0.f, 0.f};
    float sumsq = 0.f;
    for (int j = 0; j < NCH; ++j) {
        v2f a = *(const v2f*)(xrow + 4 * j);
        v2f b = *(const v2f*)(&bop[(j * 32 + lane) * 2]);
        c = __builtin_amdgcn_wmma_f32_16x16x4_f32(
                false, a, false, b, (short)0, c, false, false);
        sumsq = fmaf(a.x, a.x, fmaf(a.y, a.y, sumsq));   // lane-half partial of sum(x^2)
    }
    // combine the two lane-halves of sum(x^2) per token
    sumsq += __shfl_xor(sumsq, 16, 32);

    // C spill: VGPR v at lane -> C[m = v + 8*half][n = lm]
#pragma unroll
    for (int v = 0; v < 8; ++v) csc[wave][v + 8 * half][lm] = c[v];
    __syncthreads();

    // ----- Phase 2: per-token LN fixup + softmax (lanes 0..15, token = lane) -----
    if (lane < 16) {
        float dt[KSLOT];
#pragma unroll
        for (int k = 0; k < KSLOT; ++k) dt[k] = csc[wave][lane][k];
        const float sumx = csc[wave][lane][KSLOT];      // ones-column result
        const float mu   = sumx * (1.f / DDIM);
        const float var  = sumsq * (1.f / DDIM) - mu * mu;
        const float inv  = rsqrtf(var + 1e-5f);
        float s[KSLOT];
#pragma unroll
        for (int k = 0; k < KSLOT; ++k)
            s[k] = inv * (dt[k] - mu * wscal[k]) + wscal[KSLOT + k];
        float m = fmaxf(fmaxf(s[0], s[1]), fmaxf(s[2], s[3]));
        float e[KSLOT], tot = 0.f;
#pragma unroll
        for (int k = 0; k < KSLOT; ++k) { e[k] = __expf(s[k] - m); tot += e[k]; }
        const float rinv = 1.f / tot;
#pragma unroll
        for (int k = 0; k < KSLOT; ++k) pbuf[wave][lane][k] = e[k] * rinv;
    }
    __syncthreads();

    // ----- Phase 3: acc[k][d] += p[t][k] * x[t][d], register-resident -----
    // Lane owns d-positions { j*128 + 4*lane + c }. Second x read hits L2
    // (whole x = 100 MB fits in the 192 MB L2).
    v4f acc[KSLOT][6];
#pragma unroll
    for (int k = 0; k < KSLOT; ++k)
#pragma unroll
        for (int j = 0; j < 6; ++j) { v4f z = {0.f, 0.f, 0.f, 0.f}; acc[k][j] = z; }

    for (int t = 0; t < 16; ++t) {
        const float* xr = x + (tok0 + t) * DDIM + 4 * lane;
        const float p0 = pbuf[wave][t][0];
        const float p1 = pbuf[wave][t][1];
        const float p2 = pbuf[wave][t][2];
        const float p3 = pbuf[wave][t][3];
#pragma unroll
        for (int j = 0; j < 6; ++j) {
            v4f xv = *(const v4f*)(xr + j * 128);
            acc[0][j] += p0 * xv;
            acc[1][j] += p1 * xv;
            acc[2][j] += p2 * xv;
            acc[3][j] += p3 * xv;
        }
    }

    // wave -> block accumulator (LDS float atomics: ds_add_f32)
#pragma unroll
    for (int k = 0; k < KSLOT; ++k)
#pragma unroll
        for (int j = 0; j < 6; ++j)
#pragma unroll
            for (int cc = 0; cc < 4; ++cc)
                atomicAdd(&lacc[k * DDIM + j * 128 + 4 * lane + cc], acc[k][j][cc]);
    __syncthreads();

    // block -> global (all 128 tokens of this block share one batch: 128 | 4096)
    const int batch = (int)(tok0 >> 12);
    float* ob = out + batch * (KSLOT * DDIM);
    for (int i = tid; i < KSLOT * DDIM; i += 256) atomicAdd(&ob[i], lacc[i]);
}

// ---------------------------------------------------------------------------
extern "C" void kernel_launch(void* const* d_in, const int* in_sizes, int n_in,
                              void* d_out, int out_size, void* d_ws, size_t ws_size,
                              hipStream_t stream) {
    const float* x    = (const float*)d_in[0];   // (8,4096,768)
    const float* sk   = (const float*)d_in[1];   // (1,4,768)
    const float* sv   = (const float*)d_in[2];   // (1,4,768)
    const float* in_w = (const float*)d_in[3];   // (768,)
    const float* in_b = (const float*)d_in[4];
    const float* sl_w = (const float*)d_in[5];
    const float* sl_b = (const float*)d_in[6];
    float* out = (float*)d_out;                  // (8,4,768)

    float* wsb   = (float*)d_ws;                 // 192*64 floats = 48 KB
    float* wscal = wsb + NCH * 64;               // 8 floats

    prep_kernel<<<1, 256, 0, stream>>>(sk, in_w, in_b, sl_w, sl_b, wsb, wscal);
    init_out_kernel<<<(BATCH * KSLOT * DDIM) / 256, 256, 0, stream>>>(sv, out);
    fused_kernel<<<(BATCH * NTOK) / 128, 256, 0, stream>>>(x, wsb, wscal, out);
}